// MultiHeadedAttention_44530220925450
// MI455X (gfx1250) — compile-verified
//
#include <hip/hip_runtime.h>
#include <math.h>

typedef float v2f __attribute__((ext_vector_type(2)));
typedef float v8f __attribute__((ext_vector_type(8)));

#define MODEL_D 512
#define LQ 128
#define LKV 512
#define NBATCH 4

// ---------------------------------------------------------------------------
// fp32 WMMA GEMM:  C[M,N] = (A[M,K] @ W[K,N] + bias[N]) * alpha
// One wave computes a 16x64 output block (4x 16x16 WMMA accumulators),
// looping K in steps of 4 using v_wmma_f32_16x16x4_f32.
// A-fragment layout (ISA 7.12.2, 32-bit A 16x4): lane = 16*(K-half)+M,
// vgpr0 = K even-of-pair, vgpr1 = K odd-of-pair. B symmetric (4x16).
// C/D: vgpr r -> row ( (lane>>4)*8 + r ), col (lane&15).
// ---------------------------------------------------------------------------
__global__ __launch_bounds__(256)
void gemm_bias_wmma(const float* __restrict__ A, const float* __restrict__ W,
                    const float* __restrict__ bias, float alpha,
                    float* __restrict__ C, int M, int K, int N) {
  const int lane = threadIdx.x & 31;
  const int wave = threadIdx.x >> 5;
  const int gwave = blockIdx.x * (blockDim.x >> 5) + wave;
  const int tilesN = N >> 6;                 // 64-wide N blocks
  const int tM = gwave / tilesN;
  const int tN = gwave - tM * tilesN;
  if (tM * 16 >= M) return;                  // uniform per wave (EXEC stays all-ones)

  const int m0 = tM * 16;
  const int n0 = tN * 64;
  const int mA = lane & 15;                  // A row within tile
  const int kHalf = (lane >> 4) << 1;        // 0 or 2 (K pair base)
  const int nB = lane & 15;                  // B/C column within 16-wide subtile

  const float* __restrict__ Arow = A + (size_t)(m0 + mA) * K;

  v8f acc0 = {}, acc1 = {}, acc2 = {}, acc3 = {};

  for (int k0 = 0; k0 < K; k0 += 4) {
    const int ka = k0 + kHalf;
    v2f a;                                   // contiguous 8B -> global_load_b64
    a.x = Arow[ka];
    a.y = Arow[ka + 1];

    const float* __restrict__ Wr0 = W + (size_t)ka * N + n0 + nB;
    const float* __restrict__ Wr1 = Wr0 + N;
    v2f b0, b1, b2, b3;
    b0.x = Wr0[0];   b0.y = Wr1[0];
    b1.x = Wr0[16];  b1.y = Wr1[16];
    b2.x = Wr0[32];  b2.y = Wr1[32];
    b3.x = Wr0[48];  b3.y = Wr1[48];

    acc0 = __builtin_amdgcn_wmma_f32_16x16x4_f32(false, a, false, b0, (short)0, acc0, false, false);
    acc1 = __builtin_amdgcn_wmma_f32_16x16x4_f32(false, a, false, b1, (short)0, acc1, false, false);
    acc2 = __builtin_amdgcn_wmma_f32_16x16x4_f32(false, a, false, b2, (short)0, acc2, false, false);
    acc3 = __builtin_amdgcn_wmma_f32_16x16x4_f32(false, a, false, b3, (short)0, acc3, false, false);
  }

  const float bb0 = bias ? bias[n0 + nB]      : 0.0f;
  const float bb1 = bias ? bias[n0 + nB + 16] : 0.0f;
  const float bb2 = bias ? bias[n0 + nB + 32] : 0.0f;
  const float bb3 = bias ? bias[n0 + nB + 48] : 0.0f;

  const int mBase = (lane >> 4) * 8;
#pragma unroll
  for (int r = 0; r < 8; ++r) {
    const int m = m0 + mBase + r;
    float* __restrict__ crow = C + (size_t)m * N + n0 + nB;
    crow[0]  = (acc0[r] + bb0) * alpha;
    crow[16] = (acc1[r] + bb1) * alpha;
    crow[32] = (acc2[r] + bb2) * alpha;
    crow[48] = (acc3[r] + bb3) * alpha;
  }
}

// ---------------------------------------------------------------------------
// tanh: UNCONDITIONAL gfx1250 builtin -> v_tanh_f32 (TRANS op, co-executes
// with the FMA stream). Compile-probe: if this name is wrong the frontend
// error will say so and we switch spelling next round.
// ---------------------------------------------------------------------------
__device__ __forceinline__ float dtanh(float x) {
  return __builtin_amdgcn_tanhf(x);
}

// ---------------------------------------------------------------------------
// Fused additive-attention scoring + log_softmax.
// One 256-thread (8-wave) workgroup per (b, i) query row.
//   logits[j] = sum_d tanh(qp[row,d] + bc[d] + kp[b,j,d]) * wl[d] + bl + bias(mask)
//   out[row,:] = logits - logsumexp(logits)
// qp[row]+bc and wl staged in LDS (read back as b128); kp rows streamed
// contiguously as float4 (kp[b] is 1 MB -> L2-resident across the batch).
// ---------------------------------------------------------------------------
__global__ __launch_bounds__(256)
void attn_logsoftmax(const float* __restrict__ qp, const float* __restrict__ kp,
                     const float* __restrict__ bc, const float* __restrict__ wl,
                     const float* __restrict__ blp, const int* __restrict__ mask,
                     float* __restrict__ out) {
  __shared__ __align__(16) float qb[MODEL_D];
  __shared__ __align__(16) float wls[MODEL_D];
  __shared__ float logits[LKV];
  __shared__ float red[256];

  const int tid = threadIdx.x;
  const int row = blockIdx.x;                // b*LQ + i
  const int b = row >> 7;                    // LQ == 128

  const float* __restrict__ qrow = qp + (size_t)row * MODEL_D;
  for (int d = tid; d < MODEL_D; d += 256) {
    qb[d] = qrow[d] + bc[d];
    wls[d] = wl[d];
  }
  __syncthreads();

  const float bl = blp[0];

  for (int jj = tid; jj < LKV; jj += 256) {
    const float* __restrict__ krow = kp + ((size_t)(b * LKV + jj)) * MODEL_D;
    float acc = 0.0f;
#pragma unroll 2
    for (int d = 0; d < MODEL_D; d += 4) {
      const float4 kv = *reinterpret_cast<const float4*>(krow + d);
      const float4 qv = *reinterpret_cast<const float4*>(&qb[d]);   // ds_load_b128
      const float4 wv = *reinterpret_cast<const float4*>(&wls[d]);  // ds_load_b128
      acc += dtanh(qv.x + kv.x) * wv.x;
      acc += dtanh(qv.y + kv.y) * wv.y;
      acc += dtanh(qv.z + kv.z) * wv.z;
      acc += dtanh(qv.w + kv.w) * wv.w;
    }
    const float biasv = (mask[b * LKV + jj] == 0) ? -1.0e9f : 1.0f;
    logits[jj] = acc + bl + biasv;
  }
  __syncthreads();

  // row max
  float v = fmaxf(logits[tid], logits[tid + 256]);
  red[tid] = v;
  __syncthreads();
  for (int s = 128; s > 0; s >>= 1) {
    if (tid < s) red[tid] = fmaxf(red[tid], red[tid + s]);
    __syncthreads();
  }
  const float rmax = red[0];
  __syncthreads();

  // sum of exp
  float e = __expf(logits[tid] - rmax) + __expf(logits[tid + 256] - rmax);
  red[tid] = e;
  __syncthreads();
  for (int s = 128; s > 0; s >>= 1) {
    if (tid < s) red[tid] += red[tid + s];
    __syncthreads();
  }
  const float lse = rmax + __logf(red[0]);

  float* __restrict__ orow = out + (size_t)row * LKV;
  orow[tid]       = logits[tid] - lse;
  orow[tid + 256] = logits[tid + 256] - lse;
}

// ---------------------------------------------------------------------------
// Launch: 4 WMMA projection GEMMs (v projection is dead code -> skipped),
// then the fused scoring kernel. All on `stream`, scratch in d_ws.
// ---------------------------------------------------------------------------
extern "C" void kernel_launch(void* const* d_in, const int* in_sizes, int n_in,
                              void* d_out, int out_size, void* d_ws, size_t ws_size,
                              hipStream_t stream) {
  (void)in_sizes; (void)n_in; (void)out_size; (void)ws_size;

  const float* key   = (const float*)d_in[0];
  // d_in[1] = value (unused: v projection never reaches the output)
  const float* query = (const float*)d_in[2];
  const int*   mask  = (const int*)d_in[3];
  const float* Wk    = (const float*)d_in[4];
  const float* bk    = (const float*)d_in[5];
  // d_in[6] = Wv, d_in[7] = bv (unused)
  const float* Wq    = (const float*)d_in[8];
  const float* bq    = (const float*)d_in[9];
  const float* Wc    = (const float*)d_in[10];   // [2D, D]
  const float* bc    = (const float*)d_in[11];
  const float* wl    = (const float*)d_in[12];
  const float* bl    = (const float*)d_in[13];   // scalar as 1-elem array
  float* out = (float*)d_out;

  const int D = MODEL_D;
  const int MK = NBATCH * LKV;   // 2048 flattened key rows
  const int MQ = NBATCH * LQ;    // 512 flattened query rows

  float* kbuf = (float*)d_ws;              // [2048, 512]
  float* qbuf = kbuf + (size_t)MK * D;     // [512, 512]
  float* kp   = qbuf + (size_t)MQ * D;     // [2048, 512]
  float* qp   = kp   + (size_t)MK * D;     // [512, 512]

  const float invSqrtD = 1.0f / sqrtf((float)D);

  // wave-tiles = (M/16)*(N/64); 8 waves per 256-thread block
  const int blkK = ((MK / 16) * (D / 64)) / 8;   // 128 blocks
  const int blkQ = ((MQ / 16) * (D / 64)) / 8;   // 32 blocks

  gemm_bias_wmma<<<blkK, 256, 0, stream>>>(key,   Wk, bk, 1.0f,     kbuf, MK, D, D);
  gemm_bias_wmma<<<blkQ, 256, 0, stream>>>(query, Wq, bq, invSqrtD, qbuf, MQ, D, D);
  gemm_bias_wmma<<<blkK, 256, 0, stream>>>(kbuf,  Wc + (size_t)D * D, nullptr, 1.0f, kp, MK, D, D);
  gemm_bias_wmma<<<blkQ, 256, 0, stream>>>(qbuf,  Wc,                 nullptr, 1.0f, qp, MQ, D, D);

  attn_logsoftmax<<<NBATCH * LQ, 256, 0, stream>>>(qp, kp, bc, wl, bl, mask, out);
}